// SIURec_59923383714544
// MI455X (gfx1250) — compile-verified
//
#include <hip/hip_runtime.h>

typedef __attribute__((ext_vector_type(2))) float v2f;
typedef __attribute__((ext_vector_type(8))) float v8f;

// ---------------------------------------------------------------------------
// init: cur = acc = concat(user_emb, item_emb)   (acc lives in d_out[0:N*D])
// ---------------------------------------------------------------------------
__global__ __launch_bounds__(256) void k_init(const float* __restrict__ ue,
                                              const float* __restrict__ ie,
                                              float* __restrict__ cur,
                                              float* __restrict__ acc,
                                              int totalV4, int uV4) {
    int i = blockIdx.x * blockDim.x + threadIdx.x;
    if (i >= totalV4) return;
    float4 v = (i < uV4) ? ((const float4*)ue)[i] : ((const float4*)ie)[i - uV4];
    ((float4*)cur)[i] = v;
    ((float4*)acc)[i] = v;
}

// deg[r] += 1 per edge (int atomics, once per call)
__global__ __launch_bounds__(256) void k_deg(const int* __restrict__ rows,
                                             int* __restrict__ deg, int E_) {
    int e = blockIdx.x * blockDim.x + threadIdx.x;
    if (e < E_) atomicAdd(&deg[rows[e]], 1);
}

// single-block exclusive scan of deg -> rowptr (and a scatter-counter copy)
__global__ __launch_bounds__(1024) void k_scan(const int* __restrict__ deg,
                                               int* __restrict__ rowptr,
                                               int* __restrict__ sctr, int n) {
    __shared__ int sh[1024];
    __shared__ int carry;
    int t = threadIdx.x;
    if (t == 0) carry = 0;
    __syncthreads();
    for (int base = 0; base < n; base += 1024) {
        int i = base + t;
        int v = (i < n) ? deg[i] : 0;
        sh[t] = v;
        __syncthreads();
        for (int off = 1; off < 1024; off <<= 1) {
            int x = (t >= off) ? sh[t - off] : 0;
            __syncthreads();
            sh[t] += x;
            __syncthreads();
        }
        int incl = sh[t];
        int excl = incl - v + carry;   // reads carry before it is updated
        if (i < n) { rowptr[i] = excl; sctr[i] = excl; }
        __syncthreads();
        if (t == 1023) carry += incl;  // incl of last thread == chunk total
        __syncthreads();
    }
}

// counting-sort scatter: edge e -> slot in its row's CSR segment
__global__ __launch_bounds__(256) void k_scatter(const int* __restrict__ rows,
                                                 const int* __restrict__ cols,
                                                 int* __restrict__ sctr,
                                                 int* __restrict__ col_s,
                                                 int* __restrict__ rows_s, int E_) {
    int e = blockIdx.x * blockDim.x + threadIdx.x;
    if (e >= E_) return;
    int r = rows[e];
    int p = atomicAdd(&sctr[r], 1);
    col_s[p] = cols[e];
    rows_s[p] = r;
}

// d_is[n] = deg>0 ? deg^-0.5 : 0
__global__ __launch_bounds__(256) void k_dis(const int* __restrict__ deg,
                                             float* __restrict__ d_is, int n) {
    int i = blockIdx.x * blockDim.x + threadIdx.x;
    if (i >= n) return;
    int d = deg[i];
    d_is[i] = (d > 0) ? 1.0f / sqrtf((float)d) : 0.0f;
}

// g_vals (in CSR order) = d_is[r] * d_is[c]   (static across layers)
__global__ __launch_bounds__(256) void k_gvals(const int* __restrict__ rows_s,
                                               const int* __restrict__ col_s,
                                               const float* __restrict__ d_is,
                                               float* __restrict__ gvals, int E_) {
    int p = blockIdx.x * blockDim.x + threadIdx.x;
    if (p < E_) gvals[p] = d_is[rows_s[p]] * d_is[col_s[p]];
}

// ---------------------------------------------------------------------------
// SpMM #1 (atomic-free, wave-per-row): gnn[r] = sum_e g * cur[col]
// fused epilogue: inv_norm[r] = 1 / max(||gnn[r]||, eps)
// ---------------------------------------------------------------------------
__global__ __launch_bounds__(256) void k_gnn(const float* __restrict__ cur,
                                             float* __restrict__ gnn,
                                             float* __restrict__ inv_norm,
                                             const int* __restrict__ rowptr,
                                             const int* __restrict__ deg,
                                             const int* __restrict__ col_s,
                                             const float* __restrict__ gvals, int n) {
    int wid  = (blockIdx.x * blockDim.x + threadIdx.x) >> 5;
    int lane = threadIdx.x & 31;
    if (wid >= n) return;
    int start = rowptr[wid], cnt = deg[wid];
    const float2* curb = (const float2*)cur;
    float2 acc = make_float2(0.f, 0.f);
    for (int i = 0; i < cnt; ++i) {
        int p = start + i;
        int c = col_s[p];
        float g = gvals[p];
        float2 x = curb[(size_t)c * 32 + lane];
        acc.x = fmaf(g, x.x, acc.x);
        acc.y = fmaf(g, x.y, acc.y);
    }
    ((float2*)gnn)[(size_t)wid * 32 + lane] = acc;
    float s = acc.x * acc.x + acc.y * acc.y;
    for (int off = 16; off; off >>= 1) s += __shfl_xor(s, off, 32);
    if (lane == 0) inv_norm[wid] = 1.0f / fmaxf(sqrtf(s), 1e-12f);
}

// ---------------------------------------------------------------------------
// per-edge cosine score via V_WMMA_F32_16X16X4_F32 trace trick (wave-per-edge)
//   A[m][k] = gr[4m+k]  (ISA f32 16x4 A layout: lanes0-15 K={0,1}, lanes16-31 K={2,3})
//   B[k][n] = gc[4n+k]  (mirrored layout -> identical per-lane addressing)
//   diag(C)[i] = sum_k gr[4i+k]*gc[4i+k];  trace(C) = dot(gr, gc)
// ---------------------------------------------------------------------------
__global__ __launch_bounds__(256) void k_scores(const float* __restrict__ gnn,
                                                const float* __restrict__ inv_norm,
                                                const int* __restrict__ rows_s,
                                                const int* __restrict__ col_s,
                                                float* __restrict__ scores, int E_) {
    int wid  = (blockIdx.x * blockDim.x + threadIdx.x) >> 5;
    int lane = threadIdx.x & 31;
    int p = (wid < E_) ? wid : (E_ - 1);       // clamp: keep EXEC all-ones for WMMA
    int r = rows_s[p];
    int c = col_s[p];
    const float* gr = gnn + (size_t)r * 64;
    const float* gc = gnn + (size_t)c * 64;
    int m = lane & 15;
    int h = lane >> 4;
    int off = m * 4 + h * 2;
    v2f a; a.x = gr[off]; a.y = gr[off + 1];
    v2f b; b.x = gc[off]; b.y = gc[off + 1];
    v8f cz = {0.f, 0.f, 0.f, 0.f, 0.f, 0.f, 0.f, 0.f};
    v8f d = __builtin_amdgcn_wmma_f32_16x16x4_f32(false, a, false, b,
                                                  (short)0, cz, false, false);
    // extract diagonal: (i,i) lives at vgpr i, lane i (i<8) and vgpr i-8, lane i+16 (i>=8)
    float diag = 0.f;
#pragma unroll
    for (int k = 0; k < 8; ++k) {
        diag = (lane == k)      ? d[k] : diag;
        diag = (lane == 24 + k) ? d[k] : diag;
    }
    for (int o = 16; o; o >>= 1) diag += __shfl_xor(diag, o, 32);
    float s = fmaf(diag * inv_norm[r] * inv_norm[c], 0.5f, 0.5f);
    if (lane == 0 && wid < E_) scores[p] = s;
}

// rowsum of scores per row, inverted: d_invs[r] = rowsum>0 ? 1/rowsum : 0
__global__ __launch_bounds__(256) void k_rowsum(const float* __restrict__ scores,
                                                const int* __restrict__ rowptr,
                                                const int* __restrict__ deg,
                                                float* __restrict__ d_invs, int n) {
    int i = blockIdx.x * blockDim.x + threadIdx.x;
    if (i >= n) return;
    int s0 = rowptr[i], c = deg[i];
    float s = 0.f;
    for (int k = 0; k < c; ++k) s += scores[s0 + k];
    d_invs[i] = (s > 0.f) ? 1.0f / s : 0.0f;
}

// ---------------------------------------------------------------------------
// SpMM #2 fused with layer update (wave-per-row):
//   f = sum_e (d_inv[r]*score) * cur[col];  cur_next[r] = gnn[r] + f (in place);
//   acc(d_out)[r] += cur_next[r];           fine -> user_fine/item_fine output
// ---------------------------------------------------------------------------
__global__ __launch_bounds__(256) void k_fine(const float* __restrict__ cur,
                                              float* __restrict__ gnn_cur,
                                              const float* __restrict__ scores,
                                              const float* __restrict__ d_invs,
                                              const int* __restrict__ rowptr,
                                              const int* __restrict__ deg,
                                              const int* __restrict__ col_s,
                                              float* __restrict__ out,
                                              int U_, int I_, int layer, int n) {
    int wid  = (blockIdx.x * blockDim.x + threadIdx.x) >> 5;
    int lane = threadIdx.x & 31;
    if (wid >= n) return;
    int start = rowptr[wid], cnt = deg[wid];
    float di = d_invs[wid];
    const float2* curb = (const float2*)cur;
    float2 f = make_float2(0.f, 0.f);
    for (int i = 0; i < cnt; ++i) {
        int p = start + i;
        float w = di * scores[p];
        float2 x = curb[(size_t)col_s[p] * 32 + lane];
        f.x = fmaf(w, x.x, f.x);
        f.y = fmaf(w, x.y, f.y);
    }
    float2* gb = (float2*)gnn_cur + (size_t)wid * 32;
    float2 g = gb[lane];
    float2 cn = make_float2(g.x + f.x, g.y + f.y);
    gb[lane] = cn;                                  // becomes cur for next layer
    float2* ab = (float2*)out + (size_t)wid * 32;   // acc region: node-major
    float2 av = ab[lane];
    av.x += cn.x; av.y += cn.y;
    ab[lane] = av;
    size_t Nn = (size_t)(U_ + I_);
    size_t baseUF = Nn * 64;
    size_t baseIF = baseUF + (size_t)3 * U_ * 64;
    float* fo = (wid < U_)
        ? out + baseUF + ((size_t)layer * U_ + wid) * 64
        : out + baseIF + ((size_t)layer * I_ + (wid - U_)) * 64;
    ((float2*)fo)[lane] = f;
}

// ---------------------------------------------------------------------------
extern "C" void kernel_launch(void* const* d_in, const int* in_sizes, int n_in,
                              void* d_out, int out_size, void* d_ws, size_t ws_size,
                              hipStream_t stream) {
    const float* ue  = (const float*)d_in[0];
    const float* ie  = (const float*)d_in[1];
    const int*  rows = (const int*)d_in[2];
    const int*  cols = (const int*)d_in[3];
    const int U_ = in_sizes[0] / 64;
    const int I_ = in_sizes[1] / 64;
    const int E_ = in_sizes[2];
    const int Nn = U_ + I_;
    float* out = (float*)d_out;

    char* w = (char*)d_ws;
    auto alloc = [&](size_t bytes) -> void* {
        void* p = (void*)w;
        w += (bytes + 255) & ~(size_t)255;
        return p;
    };
    float* bufA     = (float*)alloc((size_t)Nn * 64 * 4);
    float* bufB     = (float*)alloc((size_t)Nn * 64 * 4);
    int*   col_s    = (int*)  alloc((size_t)E_ * 4);
    int*   rows_s   = (int*)  alloc((size_t)E_ * 4);
    float* gvals    = (float*)alloc((size_t)E_ * 4);
    float* scores   = (float*)alloc((size_t)E_ * 4);
    int*   deg      = (int*)  alloc((size_t)Nn * 4);
    int*   rowptr   = (int*)  alloc((size_t)Nn * 4);
    int*   sctr     = (int*)  alloc((size_t)Nn * 4);
    float* d_is     = (float*)alloc((size_t)Nn * 4);
    float* d_invs   = (float*)alloc((size_t)Nn * 4);
    float* inv_norm = (float*)alloc((size_t)Nn * 4);

    dim3 blk(256);
    auto G = [](long long t) { return dim3((unsigned)((t + 255) / 256)); };

    const int totalV4 = Nn * 64 / 4;
    const int uV4     = U_ * 64 / 4;

    k_init<<<G(totalV4), blk, 0, stream>>>(ue, ie, bufA, out, totalV4, uV4);
    hipMemsetAsync(deg, 0, (size_t)Nn * 4, stream);
    k_deg<<<G(E_), blk, 0, stream>>>(rows, deg, E_);
    k_scan<<<1, 1024, 0, stream>>>(deg, rowptr, sctr, Nn);
    k_scatter<<<G(E_), blk, 0, stream>>>(rows, cols, sctr, col_s, rows_s, E_);
    k_dis<<<G(Nn), blk, 0, stream>>>(deg, d_is, Nn);
    k_gvals<<<G(E_), blk, 0, stream>>>(rows_s, col_s, d_is, gvals, E_);

    float* cur = bufA;
    float* oth = bufB;
    for (int l = 0; l < 3; ++l) {
        k_gnn<<<G((long long)Nn * 32), blk, 0, stream>>>(cur, oth, inv_norm,
                                                         rowptr, deg, col_s, gvals, Nn);
        k_scores<<<G((long long)E_ * 32), blk, 0, stream>>>(oth, inv_norm,
                                                            rows_s, col_s, scores, E_);
        k_rowsum<<<G(Nn), blk, 0, stream>>>(scores, rowptr, deg, d_invs, Nn);
        k_fine<<<G((long long)Nn * 32), blk, 0, stream>>>(cur, oth, scores, d_invs,
                                                          rowptr, deg, col_s, out,
                                                          U_, I_, l, Nn);
        float* t = cur; cur = oth; oth = t;
    }
}